// CuGraphSAGE_56478819942994
// MI455X (gfx1250) — compile-verified
//
#include <hip/hip_runtime.h>
#include <hip/hip_bf16.h>
#include <stdint.h>

// ---------------------------------------------------------------------------
// Problem constants (from reference: N_PER_HOP = [2048, 20480, 204800, 1024000])
// ---------------------------------------------------------------------------
#define HOP0 2048
#define HOP1 20480
#define HOP2 204800
#define HOP3 1024000

#define N0 (HOP0 + HOP1 + HOP2 + HOP3)   // 1,251,328 nodes total
#define N1 (HOP0 + HOP1 + HOP2)          //   227,328 nodes live after layer 0
#define N2 (HOP0 + HOP1)                 //    22,528 nodes live after layer 1
#define NOUT HOP0                        //     2,048 output rows

#define E0 (HOP1 + HOP2 + HOP3)          // 1,249,280 edges for layer 0
#define E1 (HOP1 + HOP2)                 //   225,280 edges for layer 1
#define E2 (HOP1)                        //    20,480 edges for layer 2

typedef __attribute__((ext_vector_type(2))) float v2f;
typedef __attribute__((ext_vector_type(8))) float v8f;

// ---------------------------------------------------------------------------
// threefry2x32 (matches JAX): 20 rounds, rotations [13,15,26,6]/[17,29,16,24]
// ---------------------------------------------------------------------------
__host__ __device__ __forceinline__ unsigned tf_rotl(unsigned x, int r) {
    return (x << r) | (x >> (32 - r));
}

__host__ __device__ inline void threefry2x32(unsigned k0, unsigned k1,
                                             unsigned c0, unsigned c1,
                                             unsigned& o0, unsigned& o1) {
    unsigned k2 = k0 ^ k1 ^ 0x1BD11BDAu;
    unsigned x0 = c0 + k0;
    unsigned x1 = c1 + k1;
#define TF4(a, b, c, d) \
    x0 += x1; x1 = tf_rotl(x1, a); x1 ^= x0; \
    x0 += x1; x1 = tf_rotl(x1, b); x1 ^= x0; \
    x0 += x1; x1 = tf_rotl(x1, c); x1 ^= x0; \
    x0 += x1; x1 = tf_rotl(x1, d); x1 ^= x0;
    TF4(13, 15, 26, 6)  x0 += k1; x1 += k2 + 1u;
    TF4(17, 29, 16, 24) x0 += k2; x1 += k0 + 2u;
    TF4(13, 15, 26, 6)  x0 += k0; x1 += k1 + 3u;
    TF4(17, 29, 16, 24) x0 += k1; x1 += k2 + 4u;
    TF4(13, 15, 26, 6)  x0 += k2; x1 += k0 + 5u;
#undef TF4
    o0 = x0; o1 = x1;
}

// dropout scale: 2.0f if kept (uniform < 0.5), else 0.0f.  Matches
// jax.random.bernoulli(fold_in(key(42), layer), 0.5, (cur_n, dim)).
__device__ __forceinline__ float dropout_scale(unsigned k0, unsigned k1,
                                               unsigned long long flat,
                                               unsigned long long size) {
    unsigned long long half = size >> 1;   // all sizes are even
    unsigned c0, c1;
    int hi;
    if (flat < half) { c0 = (unsigned)flat;          c1 = (unsigned)(flat + half); hi = 0; }
    else             { c0 = (unsigned)(flat - half); c1 = (unsigned)flat;          hi = 1; }
    unsigned x0, x1;
    threefry2x32(k0, k1, c0, c1, x0, x1);
    unsigned bits = hi ? x1 : x0;
    float u = __uint_as_float((bits >> 9) | 0x3f800000u) - 1.0f;
    return (u < 0.5f) ? 2.0f : 0.0f;
}

// ---------------------------------------------------------------------------
// Kernel 1: zero scratch region
// ---------------------------------------------------------------------------
__global__ void sage_zero_kernel(float* __restrict__ p, long long n) {
    long long i = (long long)blockIdx.x * blockDim.x + threadIdx.x;
    if (i < n) p[i] = 0.0f;
}

// ---------------------------------------------------------------------------
// Kernel 2: edge scatter  agg[col] += h[row], deg[col] += 1   (atomic f32)
// One thread handles 4 consecutive features of one edge (float4 load).
// ---------------------------------------------------------------------------
template <int DIM>
__global__ void sage_scatter_kernel(const float* __restrict__ h,
                                    const int* __restrict__ row,
                                    const int* __restrict__ col,
                                    float* __restrict__ agg,
                                    float* __restrict__ deg,
                                    int n_edges) {
    constexpr int G = DIM / 4;
    long long tid = (long long)blockIdx.x * blockDim.x + threadIdx.x;
    if (tid >= (long long)n_edges * G) return;
    int e = (int)(tid / G);
    int g = (int)(tid % G);
    int r = row[e];
    int c = col[e];
    float4 v = *(const float4*)(h + (size_t)r * DIM + g * 4);
    float* dst = agg + (size_t)c * DIM + g * 4;
    atomicAdd(dst + 0, v.x);
    atomicAdd(dst + 1, v.y);
    atomicAdd(dst + 2, v.z);
    atomicAdd(dst + 3, v.w);
    if (g == 0) atomicAdd(deg + c, 1.0f);
}

// ---------------------------------------------------------------------------
// Kernel 3: agg[n, :] /= max(deg[n], 1)
// ---------------------------------------------------------------------------
template <int DIM>
__global__ void sage_norm_kernel(float* __restrict__ agg,
                                 const float* __restrict__ deg, int n_nodes) {
    long long tid = (long long)blockIdx.x * blockDim.x + threadIdx.x;
    if (tid >= (long long)n_nodes * DIM) return;
    float d = deg[tid / DIM];
    agg[tid] = agg[tid] / fmaxf(d, 1.0f);
}

// ---------------------------------------------------------------------------
// Kernel 4: fused  out = dropout(relu(concat(agg, h) @ W^T + b))
// using V_WMMA_F32_16X16X4_F32.  One block = one 16-row node tile; the
// concat(A) tile is staged once in padded LDS and shared by all waves;
// wave w computes output-column tile [16w, 16w+16).
//
// f32 WMMA fragment layouts (ISA 7.12.2):
//   A (16x4):  lane L -> M = L&15,  VGPR{0,1} = X[M][k + 2*(L>>4) + {0,1}]
//   B (4x16):  lane L -> N = L&15,  VGPR{0,1} = W[N][k + 2*(L>>4) + {0,1}]
//   C/D:       lane L, VGPR v -> element [M = v + 8*(L>>4)][N = L&15]
// ---------------------------------------------------------------------------
template <int IN_DIM, int OUT_DIM>
__global__ __launch_bounds__(32 * (OUT_DIM / 16))
void sage_gemm_kernel(const float* __restrict__ agg,   // [>=grid*16, IN_DIM] normalized
                      const float* __restrict__ h,     // [>=grid*16, IN_DIM] self feats
                      const float* __restrict__ W,     // [OUT_DIM, 2*IN_DIM] row-major
                      const float* __restrict__ bias,  // [OUT_DIM]
                      float* __restrict__ out,         // [grid*16, OUT_DIM]
                      unsigned key0, unsigned key1,
                      unsigned long long drop_size) {
    constexpr int K   = 2 * IN_DIM;
    constexpr int LDK = K + 4;                 // pad: conflict-free ds_load_b64
    __shared__ float xs[16 * LDK];

    const int rowBase = blockIdx.x * 16;

    // Cooperative staged load of the 16 x K concat tile (coalesced on kk).
    for (int idx = threadIdx.x; idx < 16 * K; idx += blockDim.x) {
        int m  = idx / K;
        int kk = idx % K;
        int n  = rowBase + m;
        float v = (kk < IN_DIM) ? agg[(size_t)n * IN_DIM + kk]
                                : h[(size_t)n * IN_DIM + (kk - IN_DIM)];
        xs[m * LDK + kk] = v;
    }
    __syncthreads();

    const int wave    = threadIdx.x >> 5;
    const int lane    = threadIdx.x & 31;
    const int colBase = wave * 16;
    const int mj      = lane & 15;             // M for A-frag, N for B-frag
    const int khalf   = (lane >> 4) << 1;      // K sub-offset {0, 2}
    const float* wrow = W + (size_t)(colBase + mj) * K;

    v8f acc = {};
#pragma unroll
    for (int kk = 0; kk < K; kk += 4) {
        v2f a = *(const v2f*)&xs[mj * LDK + kk + khalf];
        v2f b = *(const v2f*)(wrow + kk + khalf);
        acc = __builtin_amdgcn_wmma_f32_16x16x4_f32(
            /*neg_a=*/false, a, /*neg_b=*/false, b,
            /*c_mod=*/(short)0, acc, /*reuse_a=*/false, /*reuse_b=*/false);
    }

    const int col   = colBase + mj;
    const float bc  = bias[col];
    const int mhi   = (lane >> 4) << 3;        // rows +8 for lanes 16..31
#pragma unroll
    for (int v = 0; v < 8; ++v) {
        int n = rowBase + v + mhi;
        float val = fmaxf(acc[v] + bc, 0.0f);
        unsigned long long flat = (unsigned long long)n * OUT_DIM + col;
        val *= dropout_scale(key0, key1, flat, drop_size);
        out[(size_t)n * OUT_DIM + col] = val;
    }
}

// ---------------------------------------------------------------------------
// Host launch
// ---------------------------------------------------------------------------
static inline int cdiv(long long a, int b) { return (int)((a + b - 1) / b); }

extern "C" void kernel_launch(void* const* d_in, const int* in_sizes, int n_in,
                              void* d_out, int out_size, void* d_ws, size_t ws_size,
                              hipStream_t stream) {
    const float* x    = (const float*)d_in[0];
    const int*   edge = (const int*)d_in[1];     // [2, E] row-major
    const float* W1   = (const float*)d_in[4];
    const float* b1   = (const float*)d_in[5];
    const float* W2   = (const float*)d_in[6];
    const float* b2   = (const float*)d_in[7];
    const float* W3   = (const float*)d_in[8];
    const float* b3   = (const float*)d_in[9];

    const int E_total = in_sizes[1] / 2;         // 1,249,280
    const int* src = edge;                       // edge[0]
    const int* dst = edge + E_total;             // edge[1]

    // Per-layer dropout keys: fold_in(key(42), i) == threefry2x32((0,42),(0,i))
    unsigned k0[3], k1[3];
    for (unsigned i = 0; i < 3; ++i) threefry2x32(0u, 42u, 0u, i, k0[i], k1[i]);
    const unsigned long long S0 = (unsigned long long)N0 * 128ull;  // keep-mask shapes use
    const unsigned long long S1 = (unsigned long long)N1 * 128ull;  // the reference's FULL
    const unsigned long long S2 = (unsigned long long)N2 * 64ull;   // (cur_n, dim) extent

    // Scratch layout (floats): agg (reused) | deg (reused) | h1 | h2
    float* ws  = (float*)d_ws;
    const size_t AGG_CAP = (size_t)N1 * 64;      // largest agg use
    float* agg = ws;
    float* deg = agg + AGG_CAP;                  // capacity N1
    float* h1  = deg + N1;                       // [N1, 128]
    float* h2  = h1 + (size_t)N1 * 128;          // [N2, 128]

    const int B = 256;

    // ----------------- Layer 0: in 64 (x) -> out 128 (h1, rows < N1) -------
    sage_zero_kernel<<<cdiv((long long)N1 * 64, B), B, 0, stream>>>(agg, (long long)N1 * 64);
    sage_zero_kernel<<<cdiv(N1, B), B, 0, stream>>>(deg, N1);
    sage_scatter_kernel<64><<<cdiv((long long)E0 * 16, B), B, 0, stream>>>(
        x, src, dst, agg, deg, E0);
    sage_norm_kernel<64><<<cdiv((long long)N1 * 64, B), B, 0, stream>>>(agg, deg, N1);
    sage_gemm_kernel<64, 128><<<N1 / 16, 256, 0, stream>>>(
        agg, x, W1, b1, h1, k0[0], k1[0], S0);

    // ----------------- Layer 1: in 128 (h1) -> out 128 (h2, rows < N2) -----
    sage_zero_kernel<<<cdiv((long long)N2 * 128, B), B, 0, stream>>>(agg, (long long)N2 * 128);
    sage_zero_kernel<<<cdiv(N2, B), B, 0, stream>>>(deg, N2);
    sage_scatter_kernel<128><<<cdiv((long long)E1 * 32, B), B, 0, stream>>>(
        h1, src, dst, agg, deg, E1);
    sage_norm_kernel<128><<<cdiv((long long)N2 * 128, B), B, 0, stream>>>(agg, deg, N2);
    sage_gemm_kernel<128, 128><<<N2 / 16, 256, 0, stream>>>(
        agg, h1, W2, b2, h2, k0[1], k1[1], S1);

    // ----------------- Layer 2: in 128 (h2) -> out 64 (d_out, rows < 2048) -
    sage_zero_kernel<<<cdiv((long long)NOUT * 128, B), B, 0, stream>>>(agg, (long long)NOUT * 128);
    sage_zero_kernel<<<cdiv(NOUT, B), B, 0, stream>>>(deg, NOUT);
    sage_scatter_kernel<128><<<cdiv((long long)E2 * 32, B), B, 0, stream>>>(
        h2, src, dst, agg, deg, E2);
    sage_norm_kernel<128><<<cdiv((long long)NOUT * 128, B), B, 0, stream>>>(agg, deg, NOUT);
    sage_gemm_kernel<128, 64><<<NOUT / 16, 128, 0, stream>>>(
        agg, h2, W3, b3, (float*)d_out, k0[2], k1[2], S2);
}